// SWD15_28449863369559
// MI455X (gfx1250) — compile-verified
//
#include <hip/hip_runtime.h>
#include <hip/hip_bf16.h>
#include <stdint.h>

// Problem constants (fixed by reference setup_inputs)
constexpr int B = 8;
constexpr int L = 4096;   // sequence length (sorted dimension)
constexpr int D = 512;    // channels
constexpr int TILE_C = 16;         // channels per block in kernel B
constexpr int ROWSTRIDE = TILE_C + 1;  // TDM-padded LDS row stride (DWORDs), 17 coprime 64

typedef __attribute__((ext_vector_type(4))) unsigned u32x4;
typedef __attribute__((ext_vector_type(8))) unsigned u32x8;

// ---- float <-> order-preserving uint32 ------------------------------------
__device__ __forceinline__ unsigned f2s(float f) {
  unsigned u = __float_as_uint(f);
  return (u & 0x80000000u) ? ~u : (u | 0x80000000u);
}
__device__ __forceinline__ float s2f(unsigned s) {
  unsigned u = (s & 0x80000000u) ? (s & 0x7fffffffu) : ~s;
  return __uint_as_float(u);
}

// ---- CDNA5 async global->LDS loads (ASYNCcnt path) ------------------------
__device__ __forceinline__ void async_load_b128(unsigned ldsAddr, const void* gaddr) {
  asm volatile("global_load_async_to_lds_b128 %0, %1, off"
               :: "v"(ldsAddr), "v"(gaddr) : "memory");
}
__device__ __forceinline__ void wait_async0() {
  asm volatile("s_wait_asynccnt 0x0" ::: "memory");
}

// ---- CDNA5 Tensor Data Mover: one-instruction strided tile DMA to LDS -----
// D# groups in consecutive SGPRs; groups 2/3 zero-filled (tile dims 2..4 = 0).
__device__ __forceinline__ void tdm_load_tile_to_lds(u32x4 g0, u32x8 g1) {
  u32x4 z = {0u, 0u, 0u, 0u};
  asm volatile("tensor_load_to_lds %0, %1, %2, %3"
               :: "s"(g0), "s"(g1), "s"(z), "s"(z) : "memory");
}

// ===========================================================================
// Kernel A: per batch, stable sort of channel 0 via 64-bit (value,index) keys.
// Produces rank0 (inverse permutation) in workspace and cls index in d_out.
// ===========================================================================
__global__ __launch_bounds__(1024) void SWD15_rank_kernel(
    const float* __restrict__ v, unsigned* __restrict__ rank_ws,
    float* __restrict__ cls_out) {
  __shared__ unsigned long long keys[L];  // 32 KB
  const int b = blockIdx.x;
  const int t = threadIdx.x;
  const size_t base = (size_t)b * L * D;

  for (int i = t; i < L; i += 1024) {
    unsigned s = f2s(v[base + (size_t)i * D]);               // channel 0
    keys[i] = ((unsigned long long)s << 32) | (unsigned)i;   // stable key
  }
  __syncthreads();

  for (int k = 2; k <= L; k <<= 1) {
    for (int j = k >> 1; j > 0; j >>= 1) {
      for (int i = t; i < L; i += 1024) {
        int ixj = i ^ j;
        if (ixj > i) {
          unsigned long long a = keys[i], c = keys[ixj];
          if ((a > c) == ((i & k) == 0)) { keys[i] = c; keys[ixj] = a; }
        }
      }
      __syncthreads();
    }
  }

  for (int j = t; j < L; j += 1024) {
    unsigned orig = (unsigned)(keys[j] & 0xffffffffu);
    rank_ws[(size_t)b * L + orig] = (unsigned)j;
  }
  if (t == 0) cls_out[b] = (float)(unsigned)(keys[0] & 0xffffffffu);
}

// ===========================================================================
// Kernel B: one block = 16 channels x 4096 rows.
//   - TDM tensor_load_to_lds DMAs the strided tile, padding rows to 17 DWORDs
//   - ranks via async b128 loads (ASYNCcnt)
//   - bitonic sort 16 columns in LDS (order-preserving uint32)
//   - gather at rank0[i], coalesced 64B writes to both output copies
// ===========================================================================
__global__ __launch_bounds__(1024) void SWD15_sort_gather_kernel(
    const float* __restrict__ v, const unsigned* __restrict__ rank_ws,
    float* __restrict__ out) {
  __shared__ __align__(16) unsigned cols[L * ROWSTRIDE];  // 272 KB, elem(i,c)=cols[i*17+c]
  __shared__ __align__(16) unsigned ranks[L];             // 16 KB

  const int b  = blockIdx.x / (D / TILE_C);
  const int cg = blockIdx.x % (D / TILE_C);
  const int d0 = cg * TILE_C;
  const int t  = threadIdx.x;

  const unsigned colsBase  = (unsigned)(size_t)&cols[0];
  const unsigned ranksBase = (unsigned)(size_t)&ranks[0];
  const size_t vbase = (size_t)b * L * D + (size_t)d0;

  // ---- Phase 1a: ranks for this batch, 16B/lane async copies -------------
  {
    const unsigned* g = rank_ws + (size_t)b * L + (size_t)t * 4;
    async_load_b128(ranksBase + 16u * (unsigned)t, g);
  }

  // ---- Phase 1b: TDM DMA of the 16x4096 tile (wave 0 issues) -------------
  if (t < 32) {
    unsigned long long ga = (unsigned long long)(uintptr_t)(v + vbase);
    // D# group 0: count=1 | lds_addr | 57-bit global addr | type=2
    u32x4 g0 = {1u, colsBase, (unsigned)ga,
                ((unsigned)(ga >> 32) & 0x01ffffffu) | 0x80000000u};
    // D# group 1: data_size=4B, pad_enable, pad_interval=16 DW, pad_amount=1 DW,
    //             tensor_dim0=512, tensor_dim1=4096, tile_dim0=16, tile_dim1=4096,
    //             tensor_dim0_stride=512 (elements)
    u32x8 g1 = {(2u << 16) | (1u << 20) | (3u << 22) /*0x00D20000*/,
                (unsigned)(D & 0xffff) << 16,        // tensor_dim0[15:0]
                (unsigned)(L & 0xffff) << 16,        // td0[31:16]=0 | tensor_dim1[15:0]
                (unsigned)TILE_C << 16,              // td1[31:16]=0 | tile_dim0
                (unsigned)L,                         // tile_dim1 | tile_dim2=0
                (unsigned)D,                         // tensor_dim0_stride[31:0]
                0u,                                  // stride0[47:32] | stride1[15:0]
                0u};                                 // tensor_dim1_stride[47:16]
    tdm_load_tile_to_lds(g0, g1);
    __builtin_amdgcn_s_wait_tensorcnt(0);
  }
  wait_async0();
  __syncthreads();

  // ---- Phase 2a: convert in place to order-preserving uint32 -------------
  for (int s = t; s < L * TILE_C; s += 1024) {
    int c = s >> 12, i = s & (L - 1);
    unsigned idx = (unsigned)(i * ROWSTRIDE + c);
    cols[idx] = f2s(__uint_as_float(cols[idx]));
  }
  __syncthreads();

  // ---- Phase 2b: bitonic sort each of the 16 columns ----------------------
  for (int k = 2; k <= L; k <<= 1) {
    for (int j = k >> 1; j > 0; j >>= 1) {
      for (int s = t; s < L * TILE_C; s += 1024) {
        int c = s >> 12, i = s & (L - 1);
        int ixj = i ^ j;
        if (ixj > i) {
          unsigned a = cols[i * ROWSTRIDE + c];
          unsigned q = cols[ixj * ROWSTRIDE + c];
          if ((a > q) == ((i & k) == 0)) {
            cols[i * ROWSTRIDE + c] = q;
            cols[ixj * ROWSTRIDE + c] = a;
          }
        }
      }
      __syncthreads();
    }
  }

  // ---- Phase 3: gather at rank0[i], coalesced stores to both outputs -----
  float* out2 = out + (size_t)B * L * D;  // second tuple element (same data)
  for (int s = t; s < L * TILE_C; s += 1024) {
    int i = s >> 4, c = s & (TILE_C - 1);
    unsigned r = ranks[i];
    float f = s2f(cols[r * ROWSTRIDE + c]);
    size_t o = (size_t)b * L * D + (size_t)i * D + (size_t)(d0 + c);
    out[o]  = f;
    out2[o] = f;
  }
}

extern "C" void kernel_launch(void* const* d_in, const int* in_sizes, int n_in,
                              void* d_out, int out_size, void* d_ws, size_t ws_size,
                              hipStream_t stream) {
  (void)in_sizes; (void)n_in; (void)out_size; (void)ws_size;
  // inputs: q, k, v — only v is used by the reference
  const float* v = (const float*)d_in[2];
  float* out = (float*)d_out;
  unsigned* rank_ws = (unsigned*)d_ws;               // B*L*4 = 128 KB scratch
  float* cls_out = out + 2 * (size_t)B * L * D;      // tuple element 3

  SWD15_rank_kernel<<<B, 1024, 0, stream>>>(v, rank_ws, cls_out);
  SWD15_sort_gather_kernel<<<B * (D / TILE_C), 1024, 0, stream>>>(v, rank_ws, out);
}